// GNN_7318624272618
// MI455X (gfx1250) — compile-verified
//
#include <hip/hip_runtime.h>

// ---------------------------------------------------------------------------
// 2-layer GraphSAGE for MI455X (gfx1250).
// Dense GEMM phases use V_WMMA_F32_16X16X4_F32 (full fp32 MMA, 16x16 tiles,
// K=4 per op). Irregular phase uses float4 gathers + relaxed fp32 atomics.
// Memory-bound: ~1.3 GB of edge traffic -> ~56us at 23.3 TB/s; GEMMs (~2.5
// GFLOP) are negligible once on the WMMA path.
// ---------------------------------------------------------------------------

typedef __attribute__((ext_vector_type(2))) float v2f;
typedef __attribute__((ext_vector_type(8))) float v8f;

#define KDIM 64  // inner dimension of every GEMM in this net (F_IN == F_HID == 64)

__global__ void fill_zero(float* __restrict__ p, int n) {
  int i = blockIdx.x * blockDim.x + threadIdx.x;
  int stride = gridDim.x * blockDim.x;
  for (; i < n; i += stride) p[i] = 0.0f;
}

// Y[n, OUTF] = X[n, 64] @ W[OUTF, 64]^T (+ bias) (+ agg/max(cnt,1)) (ReLU?)
// One wave handles 16 rows x OUTF cols: NT = OUTF/16 accumulators, 16 K-steps.
template <int OUTF, bool RELU, bool ADD_AGG>
__global__ __launch_bounds__(256) void sage_gemm(
    const float* __restrict__ X, const float* __restrict__ W,
    const float* __restrict__ bias, const float* __restrict__ agg,
    const float* __restrict__ cnt, float* __restrict__ Y, int n) {
  constexpr int NT = OUTF / 16;
  const int wave = threadIdx.x >> 5;
  const int lane = threadIdx.x & 31;
  const int half = lane >> 4;   // 0: lanes 0-15, 1: lanes 16-31
  const int l15  = lane & 15;
  const int tile = blockIdx.x * (blockDim.x >> 5) + wave;
  const int r0 = tile * 16;
  if (r0 >= n) return;

  // A-matrix 16x4 f32 layout: lanes 0-15 hold (M=lane, K=k0+{0,1}),
  // lanes 16-31 hold (M=lane-16, K=k0+{2,3}) -> one float2 load per lane.
  int arow = r0 + l15;
  if (arow >= n) arow = n - 1;  // clamp: keep all lanes active for WMMA
  const float* xrow = X + (long)arow * KDIM + half * 2;

  // C/D init: bias depends only on column N = lane&15 -> same in all 8 regs.
  v8f acc[NT];
#pragma unroll
  for (int t = 0; t < NT; ++t) {
    float b = bias ? bias[t * 16 + l15] : 0.0f;
#pragma unroll
    for (int j = 0; j < 8; ++j) acc[t][j] = b;
  }

#pragma unroll
  for (int kb = 0; kb < KDIM / 4; ++kb) {
    v2f a = *(const v2f*)(xrow + kb * 4);
#pragma unroll
    for (int t = 0; t < NT; ++t) {
      // B-matrix 4x16 (= W^T tile): lane holds column N = lane&15;
      // B[k][N] = W[col][k] -> float2 of W row `col` at k0 + half*2.
      v2f b = *(const v2f*)(W + (t * 16 + l15) * KDIM + kb * 4 + half * 2);
      acc[t] = __builtin_amdgcn_wmma_f32_16x16x4_f32(
          /*neg_a=*/false, a, /*neg_b=*/false, b,
          /*c_mod=*/(short)0, acc[t], /*reuse_a=*/false, /*reuse_b=*/false);
    }
  }

  // D layout: VGPR v -> row r0 + v + 8*half, col = t*16 + (lane&15).
#pragma unroll
  for (int t = 0; t < NT; ++t) {
    int col = t * 16 + l15;
#pragma unroll
    for (int v = 0; v < 8; ++v) {
      int row = r0 + half * 8 + v;
      if (row < n) {
        float val = acc[t][v];
        if (ADD_AGG) {
          float c = cnt[row];
          val += agg[(long)row * OUTF + col] / fmaxf(c, 1.0f);
        }
        if (RELU) val = fmaxf(val, 0.0f);
        Y[(long)row * OUTF + col] = val;
      }
    }
  }
}

// Per (edge, float4-chunk): gather H[src] chunk, atomically add into agg[dst].
// Chunk 0 also bumps the per-dst edge count (only when cnt != nullptr).
template <int F>
__global__ __launch_bounds__(256) void sage_scatter(
    const int* __restrict__ ei, const float* __restrict__ H,
    float* __restrict__ agg, float* __restrict__ cnt, int nE) {
  constexpr int CH = F / 4;
  int idx = blockIdx.x * blockDim.x + threadIdx.x;
  if (idx >= nE * CH) return;
  int e = idx / CH;
  int c = idx - e * CH;
  int src = ei[e];        // edge_index[0][e]
  int dst = ei[nE + e];   // edge_index[1][e]
  const float4 m = *(const float4*)(H + (long)src * F + c * 4);
  float* ap = agg + (long)dst * F + c * 4;
  (void)__hip_atomic_fetch_add(ap + 0, m.x, __ATOMIC_RELAXED, __HIP_MEMORY_SCOPE_AGENT);
  (void)__hip_atomic_fetch_add(ap + 1, m.y, __ATOMIC_RELAXED, __HIP_MEMORY_SCOPE_AGENT);
  (void)__hip_atomic_fetch_add(ap + 2, m.z, __ATOMIC_RELAXED, __HIP_MEMORY_SCOPE_AGENT);
  (void)__hip_atomic_fetch_add(ap + 3, m.w, __ATOMIC_RELAXED, __HIP_MEMORY_SCOPE_AGENT);
  if (cnt != nullptr && c == 0)
    (void)__hip_atomic_fetch_add(cnt + dst, 1.0f, __ATOMIC_RELAXED, __HIP_MEMORY_SCOPE_AGENT);
}

extern "C" void kernel_launch(void* const* d_in, const int* in_sizes, int n_in,
                              void* d_out, int out_size, void* d_ws, size_t ws_size,
                              hipStream_t stream) {
  const float* x   = (const float*)d_in[0];
  const int*   ei  = (const int*)d_in[1];
  const float* Wl0 = (const float*)d_in[2];
  const float* bl0 = (const float*)d_in[3];
  const float* Wr0 = (const float*)d_in[4];
  const float* Wl1 = (const float*)d_in[5];
  const float* bl1 = (const float*)d_in[6];
  const float* Wr1 = (const float*)d_in[7];
  float* out = (float*)d_out;

  const int N = in_sizes[0] / 64;  // 100000
  const int E = in_sizes[1] / 2;   // 1600000

  // Workspace layout (floats): total N*(64*3 + 1) ~ 77 MB
  float* B0 = (float*)d_ws;           // N*64: h_lin0, later h_lin1 (N*32)
  float* B1 = B0 + (size_t)N * 64;    // N*64: agg0, later agg1 (N*32)
  float* B2 = B1 + (size_t)N * 64;    // N*64: h1 (post-ReLU layer-0 output)
  float* B3 = B2 + (size_t)N * 64;    // N   : edge counts (same both layers)

  const dim3 blk(256);
  const int tiles = (N + 15) / 16;
  const dim3 gemmGrid((tiles + 7) / 8);  // 8 waves per block

  // ---- layer 0 ----
  fill_zero<<<dim3(2048), blk, 0, stream>>>(B1, N * 64);
  fill_zero<<<dim3(256),  blk, 0, stream>>>(B3, N);
  sage_gemm<64, false, false><<<gemmGrid, blk, 0, stream>>>(x, Wl0, bl0, nullptr, nullptr, B0, N);
  {
    int work = E * 16;  // E edges * 16 float4-chunks of 64 feats
    sage_scatter<64><<<dim3((work + 255) / 256), blk, 0, stream>>>(ei, B0, B1, B3, E);
  }
  sage_gemm<64, true, true><<<gemmGrid, blk, 0, stream>>>(x, Wr0, nullptr, B1, B3, B2, N);

  // ---- layer 1 ----
  fill_zero<<<dim3(1024), blk, 0, stream>>>(B1, N * 32);
  sage_gemm<32, false, false><<<gemmGrid, blk, 0, stream>>>(B2, Wl1, bl1, nullptr, nullptr, B0, N);
  {
    int work = E * 8;   // 8 float4-chunks of 32 feats
    sage_scatter<32><<<dim3((work + 255) / 256), blk, 0, stream>>>(ei, B0, B1, nullptr, E);
  }
  sage_gemm<32, false, true><<<gemmGrid, blk, 0, stream>>>(B2, Wr1, nullptr, B1, B3, out, N);
}